// Octopus_10720238371562
// MI455X (gfx1250) — compile-verified
//
#include <hip/hip_runtime.h>
#include <hip/hip_bf16.h>

typedef __attribute__((ext_vector_type(2))) float v2f;
typedef __attribute__((ext_vector_type(8))) float v8f;

// ---------------------------------------------------------------------------
// Degree accumulation: deg[0:3N] = out-degree per relation (src side),
//                      deg[3N:6N] = in-degree per relation (dst side).
// ---------------------------------------------------------------------------
__global__ __launch_bounds__(256) void hgt_degree_kernel(
    const int* __restrict__ edges, float* __restrict__ deg, int N, int E) {
  unsigned gid = blockIdx.x * 256u + threadIdx.x;
  if (gid >= 3u * (unsigned)E) return;
  int r = gid / (unsigned)E;
  int e = gid - r * (unsigned)E;
  int src = edges[(size_t)(r * 2 + 0) * E + e];
  int dst = edges[(size_t)(r * 2 + 1) * E + e];
  atomicAdd(deg + (size_t)r * N + src, 1.0f);
  atomicAdd(deg + (size_t)(3 + r) * N + dst, 1.0f);
}

// In-place deg -> rsqrt(max(deg,1))   (DGL clamps degree to >= 1)
__global__ __launch_bounds__(256) void hgt_norm_kernel(float* __restrict__ deg,
                                                       unsigned total) {
  unsigned gid = blockIdx.x * 256u + threadIdx.x;
  if (gid >= total) return;
  deg[gid] = rsqrtf(fmaxf(deg[gid], 1.0f));
}

// ---------------------------------------------------------------------------
// Scatter: acc[r][dst] += x[src] * norm_src[r][src]  (16 threads/edge, float4)
// ---------------------------------------------------------------------------
__global__ __launch_bounds__(256) void hgt_scatter_kernel(
    const float* __restrict__ x, const int* __restrict__ edges,
    const float* __restrict__ norm_src, float* __restrict__ acc, int N, int E) {
  unsigned gid = blockIdx.x * 256u + threadIdx.x;
  if (gid >= 3u * (unsigned)E * 16u) return;
  unsigned eg = gid >> 4;
  int c = (int)(gid & 15u) << 2;
  int r = eg / (unsigned)E;
  int e = eg - r * (unsigned)E;
  int src = edges[(size_t)(r * 2 + 0) * E + e];
  int dst = edges[(size_t)(r * 2 + 1) * E + e];
  float ns = norm_src[(size_t)r * N + src];
  const float4 v = *reinterpret_cast<const float4*>(x + (size_t)src * 64 + c);
  float* ap = acc + ((size_t)r * N + dst) * 64 + c;
  atomicAdd(ap + 0, v.x * ns);
  atomicAdd(ap + 1, v.y * ns);
  atomicAdd(ap + 2, v.z * ns);
  atomicAdd(ap + 3, v.w * ns);
}

// ---------------------------------------------------------------------------
// Fused epilogue GEMM (fp32 WMMA 16x16x4):
//   out = act( sum_r (acc_r * norm_dst_r[row]) @ W_r  +  sum_r b_r )
// One wave = one 16-row node tile x all 64 output columns.
// W staged transposed in LDS so B fragments are contiguous 8B ds loads.
// ---------------------------------------------------------------------------
__global__ __launch_bounds__(128) void hgt_gemm_kernel(
    const float* __restrict__ acc,      // [3][N][64]
    const float* __restrict__ norm_in,  // [3][N]
    const float* __restrict__ W,        // [3][64][64]  (k-major, as in ref)
    const float* __restrict__ bias,     // [3][64]
    float* __restrict__ out,            // [N][64]
    int N, int ntiles, int do_relu) {
  __shared__ float sW[3 * 64 * 64];  // sW[r][n][k]  (transposed)
  __shared__ float sB[64];

  const int tid = threadIdx.x;
  for (int idx = tid; idx < 3 * 64 * 64; idx += 128) {
    int r = idx >> 12;
    int rem = idx & 4095;
    int k = rem >> 6;
    int n = rem & 63;
    sW[(r << 12) + (n << 6) + k] = W[idx];
  }
  if (tid < 64) sB[tid] = bias[tid] + bias[64 + tid] + bias[128 + tid];
  __syncthreads();

  const int wave = tid >> 5;
  const int lane = tid & 31;
  const int tile = blockIdx.x * 4 + wave;
  if (tile >= ntiles) return;
  const int row0 = tile << 4;
  const int laneM = lane & 15;          // A row within tile / B,C column
  const int hk = (lane >> 4) << 1;      // K sub-offset: 0 (lanes 0-15) or 2

  v8f cfr[4];
  cfr[0] = v8f{}; cfr[1] = v8f{}; cfr[2] = v8f{}; cfr[3] = v8f{};

  for (int r = 0; r < 3; ++r) {
    int rowA = row0 + laneM;
    if (rowA >= N) rowA = N - 1;  // N%16==0 here; clamp keeps it generic
    const float nrm = norm_in[(size_t)r * N + rowA];
    const float* arow = acc + ((size_t)r * N + rowA) * 64;

    v2f afrag[16];
#pragma unroll
    for (int ki = 0; ki < 16; ++ki) {
      v2f a = *reinterpret_cast<const v2f*>(arow + (ki << 2) + hk);
      afrag[ki] = a * nrm;  // fold norm_dst row-scaling into A fragment
    }

    const float* wr = sW + (r << 12);
#pragma unroll
    for (int nt = 0; nt < 4; ++nt) {
      const float* wcol = wr + ((nt * 16 + laneM) << 6) + hk;
      v8f c = cfr[nt];
#pragma unroll
      for (int ki = 0; ki < 16; ++ki) {
        v2f bfrag = *reinterpret_cast<const v2f*>(wcol + (ki << 2));
        c = __builtin_amdgcn_wmma_f32_16x16x4_f32(
            false, afrag[ki], false, bfrag, (short)0, c, false, false);
      }
      cfr[nt] = c;
    }
  }

  const int g = lane >> 4;
#pragma unroll
  for (int nt = 0; nt < 4; ++nt) {
    const int col = nt * 16 + laneM;
    const float bb = sB[col];
#pragma unroll
    for (int v = 0; v < 8; ++v) {
      int row = row0 + v + 8 * g;
      if (row >= N) continue;
      float val = cfr[nt][v] + bb;
      if (do_relu) val = fmaxf(val, 0.0f);
      out[(size_t)row * 64 + col] = val;
    }
  }
}

// ---------------------------------------------------------------------------
// Per-node decoder projections: ps = h2 @ Wp[0:64], pd = h2 @ Wp[64:128]
// ---------------------------------------------------------------------------
__global__ __launch_bounds__(256) void hgt_proj_kernel(
    const float* __restrict__ h2, const float* __restrict__ Wp,
    float* __restrict__ pspd, int N) {
  __shared__ float sWp[128 * 3];
  for (int i = threadIdx.x; i < 384; i += 256) sWp[i] = Wp[i];
  __syncthreads();
  int n = blockIdx.x * 256 + threadIdx.x;
  if (n >= N) return;
  const float* hr = h2 + (size_t)n * 64;
  float ps0 = 0, ps1 = 0, ps2 = 0, pd0 = 0, pd1 = 0, pd2 = 0;
#pragma unroll 8
  for (int k = 0; k < 64; ++k) {
    float h = hr[k];
    ps0 += h * sWp[k * 3 + 0];
    ps1 += h * sWp[k * 3 + 1];
    ps2 += h * sWp[k * 3 + 2];
    pd0 += h * sWp[(k + 64) * 3 + 0];
    pd1 += h * sWp[(k + 64) * 3 + 1];
    pd2 += h * sWp[(k + 64) * 3 + 2];
  }
  pspd[(size_t)n * 3 + 0] = ps0;
  pspd[(size_t)n * 3 + 1] = ps1;
  pspd[(size_t)n * 3 + 2] = ps2;
  float* pd = pspd + (size_t)3 * N;
  pd[(size_t)n * 3 + 0] = pd0;
  pd[(size_t)n * 3 + 1] = pd1;
  pd[(size_t)n * 3 + 2] = pd2;
}

// out[e] = ps[src] + pd[dst] + bp
__global__ __launch_bounds__(256) void hgt_dec_kernel(
    const int* __restrict__ dec, const float* __restrict__ pspd,
    const float* __restrict__ bp, float* __restrict__ out, int N, int ED) {
  int e = blockIdx.x * 256 + threadIdx.x;
  if (e >= ED) return;
  int s = dec[e];
  int d = dec[(size_t)ED + e];
  const float* ps = pspd + (size_t)s * 3;
  const float* pd = pspd + (size_t)3 * N + (size_t)d * 3;
  out[(size_t)e * 3 + 0] = ps[0] + pd[0] + bp[0];
  out[(size_t)e * 3 + 1] = ps[1] + pd[1] + bp[1];
  out[(size_t)e * 3 + 2] = ps[2] + pd[2] + bp[2];
}

extern "C" void kernel_launch(void* const* d_in, const int* in_sizes, int n_in,
                              void* d_out, int out_size, void* d_ws,
                              size_t ws_size, hipStream_t stream) {
  const float* x  = (const float*)d_in[0];
  const int* edges = (const int*)d_in[1];
  const int* dec   = (const int*)d_in[2];
  const float* W1 = (const float*)d_in[3];
  const float* b1 = (const float*)d_in[4];
  const float* W2 = (const float*)d_in[5];
  const float* b2 = (const float*)d_in[6];
  const float* Wp = (const float*)d_in[7];
  const float* bp = (const float*)d_in[8];
  float* out = (float*)d_out;

  const int N  = in_sizes[0] / 64;      // 100000
  const int E  = in_sizes[1] / 6;       // 800000 (3 rel x 2 x E)
  const int ED = in_sizes[2] / 2;       // 500000

  // Workspace layout (floats): norms | acc (reused both layers) | h1 | h2 | pspd
  float* ws   = (float*)d_ws;
  float* deg  = ws;                               // 6N  (-> norms in place)
  float* accb = deg + (size_t)6 * N;              // 3*N*64
  float* h1   = accb + (size_t)3 * N * 64;        // N*64
  float* h2   = h1 + (size_t)N * 64;              // N*64
  float* pspd = h2 + (size_t)N * 64;              // 6N

  const float* nsrc = deg;                        // [3][N]
  const float* ndst = deg + (size_t)3 * N;        // [3][N]

  const int ntiles = (N + 15) / 16;
  const int gblocks = (ntiles + 3) / 4;
  const unsigned degT = 3u * (unsigned)E;
  const unsigned scatT = 3u * (unsigned)E * 16u;

  // Degrees -> norms (shared by both layers)
  hipMemsetAsync(deg, 0, (size_t)6 * N * sizeof(float), stream);
  hgt_degree_kernel<<<(degT + 255) / 256, 256, 0, stream>>>(edges, deg, N, E);
  hgt_norm_kernel<<<(6u * N + 255) / 256, 256, 0, stream>>>(deg, 6u * N);

  // Layer 1: scatter x -> acc, then fused GEMM(+bias+ReLU) -> h1
  hipMemsetAsync(accb, 0, (size_t)3 * N * 64 * sizeof(float), stream);
  hgt_scatter_kernel<<<(scatT + 255) / 256, 256, 0, stream>>>(x, edges, nsrc,
                                                              accb, N, E);
  hgt_gemm_kernel<<<gblocks, 128, 0, stream>>>(accb, ndst, W1, b1, h1, N,
                                               ntiles, 1);

  // Layer 2: scatter h1 -> acc, fused GEMM(+bias) -> h2
  hipMemsetAsync(accb, 0, (size_t)3 * N * 64 * sizeof(float), stream);
  hgt_scatter_kernel<<<(scatT + 255) / 256, 256, 0, stream>>>(h1, edges, nsrc,
                                                              accb, N, E);
  hgt_gemm_kernel<<<gblocks, 128, 0, stream>>>(accb, ndst, W2, b2, h2, N,
                                               ntiles, 0);

  // Decoder: per-node projections then per-edge add
  hgt_proj_kernel<<<(N + 255) / 256, 256, 0, stream>>>(h2, Wp, pspd, N);
  hgt_dec_kernel<<<(ED + 255) / 256, 256, 0, stream>>>(dec, pspd, bp, out, N,
                                                       ED);
}